// GCNClusterP_18906446037451
// MI455X (gfx1250) — compile-verified
//
#include <hip/hip_runtime.h>

// ---------------------------------------------------------------------------
// GCN inference pipeline for MI455X (gfx1250, wave32, WMMA).
// support = h @ W   (fp32 WMMA 16x16x4, LDS-staged 64x32 block tiles)
// out[d] += w_e * support[s]  (wave-per-edge gather + f32 global atomics; the
//                              128MB activation buffer is L2-resident on the
//                              192MB L2, so atomics/gathers stay on-chip)
// relu fused into next GEMM's A-tile load; bias+relu fused into final store.
// ---------------------------------------------------------------------------

typedef float v2f __attribute__((ext_vector_type(2)));
typedef float v8f __attribute__((ext_vector_type(8)));

#define TM 64
#define TN 32
#define TK 16
#define ASTR 17   // LDS A-tile stride (16+1 to kill bank conflicts)

__global__ __launch_bounds__(256) void gcn_gemm_wmma(
    const float* __restrict__ A, int lda,
    const float* __restrict__ B, int ldb,   // packed weights [Kpad x ldb]
    float* __restrict__ C, int ldc,
    int M, int Kpad, int nValid,
    const float* __restrict__ bias, int reluA, int reluOut)
{
    __shared__ float As[TM * ASTR];
    __shared__ float Bs[TK * TN];

    const int tid  = threadIdx.x;
    const int lane = tid & 31;
    const int wave = tid >> 5;
    const int half = lane >> 4;     // 0: lanes 0-15, 1: lanes 16-31
    const int r    = lane & 15;
    const int m0   = (wave >> 1) * 16;   // 4 wave rows
    const int n0   = (wave & 1) * 16;    // 2 wave cols

    const int mBlk = blockIdx.x * TM;
    const int nBlk = blockIdx.y * TN;

    // cooperative fill mappings
    const int arow = tid >> 2;          // 0..63
    const int ac4  = (tid & 3) * 4;     // 0,4,8,12
    const int brow = tid >> 4;          // 0..15
    const int bc2  = (tid & 15) * 2;    // 0..30

    v8f c = {};

    for (int kb = 0; kb < Kpad; kb += TK) {
        // ---- stage A tile (64x16) ----
        float4 av = make_float4(0.f, 0.f, 0.f, 0.f);
        const int grow = mBlk + arow;
        if (grow < M) {
            av = *(const float4*)(A + (size_t)grow * lda + kb + ac4);
            if (reluA) {
                av.x = fmaxf(av.x, 0.f); av.y = fmaxf(av.y, 0.f);
                av.z = fmaxf(av.z, 0.f); av.w = fmaxf(av.w, 0.f);
            }
        }
        As[arow * ASTR + ac4 + 0] = av.x;
        As[arow * ASTR + ac4 + 1] = av.y;
        As[arow * ASTR + ac4 + 2] = av.z;
        As[arow * ASTR + ac4 + 3] = av.w;

        // ---- stage B tile (16x32) ----
        float2 bv = *(const float2*)(B + (size_t)(kb + brow) * ldb + nBlk + bc2);
        Bs[brow * TN + bc2 + 0] = bv.x;
        Bs[brow * TN + bc2 + 1] = bv.y;

        __syncthreads();

        // ---- 4 x V_WMMA_F32_16X16X4_F32 per K-step ----
        #pragma unroll
        for (int kk = 0; kk < TK; kk += 4) {
            v2f a, b;
            // A 16x4 layout: lanes0-15 hold K=kk,kk+1; lanes16-31 hold K=kk+2,kk+3
            a.x = As[(m0 + r) * ASTR + kk + 2 * half + 0];
            a.y = As[(m0 + r) * ASTR + kk + 2 * half + 1];
            // B 4x16 layout mirrors C: VGPR0 rows {kk | kk+2}, VGPR1 rows {kk+1 | kk+3}
            b.x = Bs[(kk + 2 * half + 0) * TN + n0 + r];
            b.y = Bs[(kk + 2 * half + 1) * TN + n0 + r];
            c = __builtin_amdgcn_wmma_f32_16x16x4_f32(
                    /*neg_a=*/false, a, /*neg_b=*/false, b,
                    /*c_mod=*/(short)0, c, /*reuse_a=*/false, /*reuse_b=*/false);
        }
        __syncthreads();
    }

    // ---- epilogue: C 16x16 layout: VGPR v -> M = v (+8 for upper half), N = lane ----
    const int col = nBlk + n0 + r;
    float bval = 0.f;
    if (bias != nullptr && col < nValid) bval = bias[col];
    #pragma unroll
    for (int v = 0; v < 8; ++v) {
        const int orow = mBlk + m0 + v + 8 * half;
        if (orow < M && col < nValid) {
            float val = c[v] + bval;
            if (reluOut) val = fmaxf(val, 0.f);
            C[(size_t)orow * ldc + col] = val;
        }
    }
}

// One wave per edge: lanes stride over float4 feature chunks of support[src],
// scatter-add into out[dst]. Both buffers are L2-resident (<=128MB each).
__global__ __launch_bounds__(256) void gcn_edge_agg(
    const int* __restrict__ src, const int* __restrict__ dst,
    const float* __restrict__ ew,
    const float* __restrict__ sup, float* __restrict__ out,
    int ld, int nvec4, int E)
{
    const int gt   = blockIdx.x * blockDim.x + threadIdx.x;
    const int e    = gt >> 5;
    const int lane = gt & 31;
    if (e >= E) return;
    const int   s = src[e];
    const int   d = dst[e];
    const float w = ew[e];
    const float4* sp = (const float4*)(sup + (size_t)s * ld);
    float*        op = out + (size_t)d * ld;
    for (int cidx = lane; cidx < nvec4; cidx += 32) {
        float4 v = sp[cidx];
        const int f = cidx * 4;
        atomicAdd(op + f + 0, w * v.x);
        atomicAdd(op + f + 1, w * v.y);
        atomicAdd(op + f + 2, w * v.z);
        atomicAdd(op + f + 3, w * v.w);
    }
}

__global__ __launch_bounds__(256) void gcn_zero4(float4* __restrict__ p, long n4)
{
    const long i = (long)blockIdx.x * blockDim.x + threadIdx.x;
    if (i < n4) p[i] = make_float4(0.f, 0.f, 0.f, 0.f);
}

// Zero-pad W [K x N] row-major into Wp [Kp x Np].
__global__ __launch_bounds__(256) void gcn_pack_w(
    const float* __restrict__ W, float* __restrict__ Wp,
    int K, int N, int Np, long total)
{
    const long i = (long)blockIdx.x * blockDim.x + threadIdx.x;
    if (i >= total) return;
    const int k = (int)(i / Np);
    const int n = (int)(i % Np);
    Wp[i] = (k < K && n < N) ? W[(long)k * N + n] : 0.f;
}

static inline long cdiv(long a, long b) { return (a + b - 1) / b; }

extern "C" void kernel_launch(void* const* d_in, const int* in_sizes, int n_in,
                              void* d_out, int out_size, void* d_ws, size_t ws_size,
                              hipStream_t stream)
{
    const float* feat = (const float*)d_in[0];
    const int*   src  = (const int*)  d_in[1];
    const int*   dst  = (const int*)  d_in[2];
    const float* ew   = (const float*)d_in[3];
    const float* W1   = (const float*)d_in[4];
    const float* W2   = (const float*)d_in[5];
    const float* W3   = (const float*)d_in[6];
    const float* Wo   = (const float*)d_in[7];
    const float* bo   = (const float*)d_in[8];

    const int IN = 256, H1 = 200, H2 = 300, H3 = 200, OUT = 100;
    // N padded to multiples of 32 (GEMM block N), K padded to multiples of 16.
    const int H1p = 224, H2p = 320, H3p = 224, OUTp = 128;
    const int K1p = 256, K2p = 208, K3p = 304, Kop = 208;
    const int LD  = 320;                 // shared leading dim for activations

    const int M = in_sizes[0] / IN;      // 100000
    const int E = in_sizes[1];           // 1600000

    // workspace layout: two ping-pong activation buffers + packed weights
    float*  ws   = (float*)d_ws;
    const size_t nb = (size_t)M * LD;
    float* bufA = ws;                                // support (GEMM out / agg in)
    float* bufB = ws + nb;                           // aggregated h (agg out / GEMM in)
    float* Wp1  = bufB + nb;
    float* Wp2  = Wp1 + (size_t)K1p * H1p;
    float* Wp3  = Wp2 + (size_t)K2p * H2p;
    float* Wpo  = Wp3 + (size_t)K3p * H3p;

    // ---- pack weights (zero-padded) ----
    {
        long t1 = (long)K1p * H1p, t2 = (long)K2p * H2p, t3 = (long)K3p * H3p, t4 = (long)Kop * OUTp;
        gcn_pack_w<<<(int)cdiv(t1, 256), 256, 0, stream>>>(W1, Wp1, IN, H1, H1p, t1);
        gcn_pack_w<<<(int)cdiv(t2, 256), 256, 0, stream>>>(W2, Wp2, H1, H2, H2p, t2);
        gcn_pack_w<<<(int)cdiv(t3, 256), 256, 0, stream>>>(W3, Wp3, H2, H3, H3p, t3);
        gcn_pack_w<<<(int)cdiv(t4, 256), 256, 0, stream>>>(Wo, Wpo, H3, OUT, OUTp, t4);
    }

    const dim3 blk(256);
    const int  gM   = (int)cdiv(M, TM);
    const int  zblk = (int)cdiv((long)(nb / 4), 256);
    const int  ablk = (int)cdiv((long)E * 32, 256);

    // ---- layer 1: support = feat_x @ W1 ; h1 = relu(scatter(support)) ----
    gcn_gemm_wmma<<<dim3(gM, H1p / TN), blk, 0, stream>>>(
        feat, IN, Wp1, H1p, bufA, LD, M, K1p, H1, nullptr, 0, 0);
    gcn_zero4<<<zblk, blk, 0, stream>>>((float4*)bufB, (long)(nb / 4));
    gcn_edge_agg<<<ablk, blk, 0, stream>>>(src, dst, ew, bufA, bufB, LD, H1 / 4, E);

    // ---- layer 2 (relu fused into A load) ----
    gcn_gemm_wmma<<<dim3(gM, H2p / TN), blk, 0, stream>>>(
        bufB, LD, Wp2, H2p, bufA, LD, M, K2p, H2, nullptr, 1, 0);
    gcn_zero4<<<zblk, blk, 0, stream>>>((float4*)bufB, (long)(nb / 4));
    gcn_edge_agg<<<ablk, blk, 0, stream>>>(src, dst, ew, bufA, bufB, LD, H2 / 4, E);

    // ---- layer 3 ----
    gcn_gemm_wmma<<<dim3(gM, H3p / TN), blk, 0, stream>>>(
        bufB, LD, Wp3, H3p, bufA, LD, M, K3p, H3, nullptr, 1, 0);
    gcn_zero4<<<zblk, blk, 0, stream>>>((float4*)bufB, (long)(nb / 4));
    gcn_edge_agg<<<ablk, blk, 0, stream>>>(src, dst, ew, bufA, bufB, LD, H3 / 4, E);

    // ---- output head: relu(h3 @ W_out + b_out) -> d_out (ldc = 100, unpadded) ----
    gcn_gemm_wmma<<<dim3(gM, OUTp / TN), blk, 0, stream>>>(
        bufB, LD, Wpo, OUTp, (float*)d_out, OUT, M, Kop, OUT, bo, 1, 1);
}